// InterpolationWeightsBase_11802570129454
// MI455X (gfx1250) — compile-verified
//
#include <hip/hip_runtime.h>
#include <cstdint>

// Bicubic grid_sample (border, align_corners=True) specialized to the
// separable outer-product grid of the reference.
//
// out[c, i, j] = sum_{a,b} wx[i,a] * wy[j,b] * v[c, yi[j,b], xi[i,a]]
//
// Roofline: output = 256 MB written once -> ~11 us floor at 23.3 TB/s HBM;
// real math is ~2.4 GFLOP (noise). So: async global->LDS scattered gathers
// (ASYNCcnt path) for the tap columns, LDS-staged separable combine, and
// non-temporal coalesced stores so the 256 MB stream does not evict the
// L2-resident 16 MB `values` tensor.
//
// Block owns (c, i0..i0+TILE_I-1); thread t == source row y.
//   1) global_load_async_to_lds_b32: v[c, y, xi[q][a]] -> tap[q][a][y]
//   2) x-combine in LDS:  Cx[q][y] = sum_a wx[q][a] * tap[q][a][y]
//   3) per output j: cubic y-weights once, 4 LDS gathers per q,
//      TILE_I coalesced NT output rows.

#define TILE_I 8
#define BLK    256   // == H (one thread per source row)

__device__ __forceinline__ void cubic_w(float t, float w[4]) {
    // Keys cubic convolution, A = -0.75 (matches PyTorch grid_sample)
    const float Af = -0.75f;
    float t2 = t * t;
    float t3 = t2 * t;
    w[0] = Af * (t3 - 2.0f * t2 + t);
    w[1] = (Af + 2.0f) * t3 - (Af + 3.0f) * t2 + 1.0f;
    float s  = 1.0f - t;
    float s2 = s * s;
    w[2] = (Af + 2.0f) * s2 * s - (Af + 3.0f) * s2 + 1.0f;
    float u = 1.0f + s;  // = 2 - t
    w[3] = Af * (u * u * u) - 5.0f * Af * (u * u) + 8.0f * Af * u - 4.0f * Af;
}

__global__ __launch_bounds__(BLK) void bicubic2d_kernel(
    const float* __restrict__ v,    // (C, H, W)
    const float* __restrict__ g0,   // (L0)  -> y coords, output dim j (innermost)
    const float* __restrict__ g1,   // (L1)  -> x coords, output dim i
    float* __restrict__ out,        // (C, L1, L0)
    int C, int H, int W, int L1, int L0)
{
    const int i0  = blockIdx.x * TILE_I;
    const int c   = blockIdx.y;
    const int tid = threadIdx.x;          // source row y (H == BLK)

    __shared__ float tap[TILE_I][4][BLK]; // 32 KB: gathered tap columns
    __shared__ float Cx[TILE_I][BLK];     //  8 KB: x-combined columns

    float wx[TILE_I][4];

    // ---- stage 1: scattered async gathers, global -> LDS (ASYNCcnt path) ----
    const float* vrow = v + ((size_t)c * H + tid) * W;   // row y of channel c
    for (int q = 0; q < TILE_I; ++q) {
        int i = i0 + q;
        float x  = (g1[i] + 1.0f) * 0.5f * (float)(W - 1);  // block-uniform -> SALU
        float xf = floorf(x);
        cubic_w(x - xf, wx[q]);
        int xb = (int)xf;
        for (int a = 0; a < 4; ++a) {
            int xa = min(max(xb - 1 + a, 0), W - 1);        // border clamp
            const float* gp = vrow + xa;
            // low 32 bits of a flat LDS pointer == LDS byte offset
            uint32_t lp = (uint32_t)(uintptr_t)&tap[q][a][tid];
            asm volatile("global_load_async_to_lds_b32 %0, %1, off"
                         :: "v"(lp), "v"(gp) : "memory");
        }
    }
    asm volatile("s_wait_asynccnt 0" ::: "memory");
    __syncthreads();

    // ---- stage 2: x-combine in LDS ----
    for (int q = 0; q < TILE_I; ++q) {
        Cx[q][tid] = wx[q][0] * tap[q][0][tid] + wx[q][1] * tap[q][1][tid]
                   + wx[q][2] * tap[q][2][tid] + wx[q][3] * tap[q][3][tid];
    }
    __syncthreads();

    // ---- stage 3: y-interp, coalesced non-temporal streaming stores ----
    for (int j = tid; j < L0; j += BLK) {
        float y  = (g0[j] + 1.0f) * 0.5f * (float)(H - 1);
        float yf = floorf(y);
        float wy[4];
        cubic_w(y - yf, wy);
        int yb = (int)yf;
        int yi[4];
        for (int b = 0; b < 4; ++b) yi[b] = min(max(yb - 1 + b, 0), H - 1);
        float* orow = out + (((size_t)c * L1) + i0) * L0 + j;
        for (int q = 0; q < TILE_I; ++q) {
            float acc = wy[0] * Cx[q][yi[0]] + wy[1] * Cx[q][yi[1]]
                      + wy[2] * Cx[q][yi[2]] + wy[3] * Cx[q][yi[3]];
            __builtin_nontemporal_store(acc, orow + (size_t)q * L0);
        }
    }
}

extern "C" void kernel_launch(void* const* d_in, const int* in_sizes, int n_in,
                              void* d_out, int out_size, void* d_ws, size_t ws_size,
                              hipStream_t stream) {
    (void)n_in; (void)out_size; (void)d_ws; (void)ws_size;
    const float* values = (const float*)d_in[0];  // (1, C, H, W) fp32
    const float* g0     = (const float*)d_in[1];  // (L0,) fp32
    const float* g1     = (const float*)d_in[2];  // (L1,) fp32
    float*       out    = (float*)d_out;          // (1, C, L1, L0) fp32

    const int H  = 256;
    const int W  = 256;
    const int L0 = in_sizes[1];
    const int L1 = in_sizes[2];
    const int C  = in_sizes[0] / (H * W);

    dim3 grid(L1 / TILE_I, C);   // (128, 64) blocks
    bicubic2d_kernel<<<grid, BLK, 0, stream>>>(values, g0, g1, out, C, H, W, L1, L0);
}